// ProtoNetwork_893353197968
// MI455X (gfx1250) — compile-verified
//
#include <hip/hip_runtime.h>

// ProtoNet episode scoring for MI455X (gfx1250, wave32).
//
//   proto  = mean_k(support)                        [B, C, H]  (kernel 1, d_ws)
//   cross  = q @ proto^T via V_WMMA_F32_16X16X4_F32 [B, CQ, C]
//   logits = -(|q|^2 + |p|^2 - 2*cross)             fused
//   pred   = argmax_c logits                        fused (shfl butterfly)
//
// Bandwidth roofline: ~84 MB HBM @ 23.3 TB/s ~ 3.6 us. Saturating 23.3 TB/s
// needs ~12 MB in flight (BW x latency); one wave per tile (480 waves) only
// sustains ~1 MB. So each tile gets a 16-wave workgroup (split-K over H:
// 128 floats/wave), partials reduced through LDS -> 7680 waves, ~15 MB of
// outstanding loads, near saturation.

typedef __attribute__((ext_vector_type(2))) float v2f;
typedef __attribute__((ext_vector_type(4))) float v4f;
typedef __attribute__((ext_vector_type(8))) float v8f;

#define B_    32
#define CW    16            // C_WAY
#define KS    5             // K_SHOT
#define QP    15            // Q_PER
#define HID   2048
#define CQ    (CW * QP)     // 240 queries per batch
#define NT    (CQ / 16)     // 15 row tiles per batch
#define WAVES 16            // split-K waves per tile
#define CHUNK (HID / WAVES) // 128 H-values per wave

// ---------------------------------------------------------------------------
// Kernel 1: class prototypes  proto[b,c,h] = mean_k support[b,c,k,h]
// ---------------------------------------------------------------------------
__global__ void proto_kernel(const float* __restrict__ support,
                             float* __restrict__ proto) {
    const int idx = blockIdx.x * blockDim.x + threadIdx.x;   // float4 index
    const int H4  = HID / 4;
    const int h4  = idx % H4;
    const int bc  = idx / H4;                                // 0 .. B*C-1
    if (bc >= B_ * CW) return;

    const v4f* s = (const v4f*)(support + (size_t)bc * KS * HID) + h4;
    v4f acc = s[0];
    #pragma unroll
    for (int k = 1; k < KS; ++k) acc += s[(size_t)k * (HID / 4)];
    acc *= 0.2f;                                             // 1/K_SHOT
    ((v4f*)(proto + (size_t)bc * HID))[h4] = acc;
}

// ---------------------------------------------------------------------------
// Kernel 2: one 16-wave workgroup per 16x16 logits tile (split-K over H).
//   A (16x4, f32): lane L -> row M = L%16, K = 2*(L/16)+{0,1}  (float2 load)
//   B (4x16, f32): lane L -> col N = L%16, K = 2*(L/16)+{0,1}  (float2 load)
//   C/D (16x16 f32, 8 VGPRs): VGPR r, lanes 0-15 -> M=r, lanes 16-31 -> M=r+8.
// Each wave accumulates a partial cross tile + partial |q|^2/|p|^2 over its
// 128-wide H chunk; partials reduced through LDS, logits + argmax fused.
// ---------------------------------------------------------------------------
__global__ void __launch_bounds__(32 * WAVES)
cross_kernel(const float* __restrict__ query,
             const float* __restrict__ proto,
             float* __restrict__ logits,
             float* __restrict__ pred) {
    __shared__ float cr[WAVES][256];   // partial cross tiles   (16 KB)
    __shared__ float lq[WAVES][16];    // partial |q_m|^2       (1 KB)
    __shared__ float lp[WAVES][16];    // partial |p_c|^2       (1 KB)

    const int tile = blockIdx.x;            // 0 .. B_*NT-1
    const int b    = tile / NT;
    const int nt   = tile % NT;
    const int tid  = threadIdx.x;
    const int w    = tid >> 5;              // wave id: K-chunk owner
    const int lane = tid & 31;
    const int m    = lane & 15;             // A row / B class col
    const int kb   = (lane >> 4) << 1;      // K base within step: 0 or 2
    const int h0   = w * CHUNK;

    const float* qrow = query + (size_t)(b * CQ + nt * 16 + m) * HID + h0;
    const float* prow = proto + (size_t)(b * CW + m) * HID + h0;

    v8f   acc  = {};
    float qacc = 0.0f;
    float pacc = 0.0f;

    #pragma unroll 8
    for (int h = 0; h < CHUNK; h += 4) {
        v2f a = *(const v2f*)(qrow + h + kb);
        v2f p = *(const v2f*)(prow + h + kb);
        // D = A x B + C   (v_wmma_f32_16x16x4_f32)
        acc = __builtin_amdgcn_wmma_f32_16x16x4_f32(
            /*neg_a=*/false, a, /*neg_b=*/false, p,
            /*c_mod=*/(short)0, acc, /*reuse_a=*/false, /*reuse_b=*/false);
        qacc += a.x * a.x + a.y * a.y;
        pacc += p.x * p.x + p.y * p.y;
    }

    // Fold the two K-halves (lanes L, L^16) -> full chunk norms per lane.
    const float qfull = qacc + __shfl_xor(qacc, 16, 32);  // |q[m]|^2 chunk
    const float pfull = pacc + __shfl_xor(pacc, 16, 32);  // |p[c]|^2 chunk

    // Spill this wave's partial tile + norms to LDS.
    const int half8 = (lane >> 4) << 3;     // 0 or 8
    #pragma unroll
    for (int r = 0; r < 8; ++r)
        cr[w][(r + half8) * 16 + m] = acc[r];
    if (lane < 16) {
        lq[w][lane] = qfull;
        lp[w][lane] = pfull;
    }
    __syncthreads();

    // Waves 0..7 (256 threads): one thread per logits element.
    if (tid < 256) {
        const int M = tid >> 4;
        const int N = tid & 15;
        float c = 0.0f, qs = 0.0f, ps = 0.0f;
        #pragma unroll
        for (int s = 0; s < WAVES; ++s) {
            c  += cr[s][tid];
            qs += lq[s][M];
            ps += lp[s][N];
        }
        const float lg  = -(qs + ps - 2.0f * c);
        const int   row = b * CQ + nt * 16 + M;
        logits[(size_t)row * CW + N] = lg;

        // Row argmax over the 16 lanes holding this row (xor<16 stays in the
        // half-wave). First-max tie-break to match jnp.argmax.
        float bv = lg;
        int   bi = N;
        #pragma unroll
        for (int d = 8; d >= 1; d >>= 1) {
            const float ov = __shfl_xor(bv, d, 32);
            const int   oi = __shfl_xor(bi, d, 32);
            if (ov > bv || (ov == bv && oi < bi)) { bv = ov; bi = oi; }
        }
        if (N == 0) pred[row] = (float)bi;
    }
}

// ---------------------------------------------------------------------------
extern "C" void kernel_launch(void* const* d_in, const int* in_sizes, int n_in,
                              void* d_out, int out_size, void* d_ws, size_t ws_size,
                              hipStream_t stream) {
    const float* support = (const float*)d_in[0];  // [B*C*K, H]
    const float* query   = (const float*)d_in[1];  // [B*C*Q, H]
    // d_in[2..4] are C, K, Q scalars — compile-time constants here.

    float* proto  = (float*)d_ws;                        // B*C*H floats = 4 MB
    float* logits = (float*)d_out;                       // B*CQ*C floats
    float* pred   = (float*)d_out + (size_t)B_ * CQ * CW;

    // proto: 262144 float4 items
    proto_kernel<<<(B_ * CW * (HID / 4) + 255) / 256, 256, 0, stream>>>(support, proto);
    // 480 tiles x 16 waves = 7680 wave32s streaming ~15 MB in flight
    cross_kernel<<<B_ * NT, 32 * WAVES, 0, stream>>>(query, proto, logits, pred);
}